// Discriminator_34746285424719
// MI455X (gfx1250) — compile-verified
//
#include <hip/hip_runtime.h>

typedef __attribute__((ext_vector_type(2))) float v2f;
typedef __attribute__((ext_vector_type(8))) float v8f;

#define TT 20
#define HH 10
#define WAVES 4
#define THREADS (WAVES * 32)

// per-wave LDS layout (floats)
#define X_OFF 0          // 32 cols * 20 x-values (recycled as score buffer) = 640
#define H_OFF 640        // 32 cols * 12 (h padded)                          = 384
#define D_OFF 1024       // 32 cols * 48 gate rows                           = 1536
#define WAVE_LDS 2560    // floats per wave (10240 B), 4 waves -> 40 KB/block

// Fast transcendentals: v_rcp_f32 / v_exp / v_tanh are TRANS32 ops that
// co-execute with the matrix pipe -- avoid the IEEE div_fmas macro-sequence.
__device__ __forceinline__ float fastrcp(float x) { return __builtin_amdgcn_rcpf(x); }
__device__ __forceinline__ float sigmf(float x)   { return fastrcp(1.0f + __expf(-x)); }
#if __has_builtin(__builtin_amdgcn_tanhf)
__device__ __forceinline__ float tanhfast(float x) { return __builtin_amdgcn_tanhf(x); }
#else
__device__ __forceinline__ float tanhfast(float x) {
    return __builtin_fmaf(2.0f, sigmf(2.0f * x), -1.0f);
}
#endif

// W_aug(48x12): rows 0..39 = [W_hh | w_x | b_ih+b_hh]; rows 40..44 = [W1 | 0 | b1]; rows 45..47 = 0
__device__ __forceinline__ float waug(int m, int k,
    const float* __restrict__ W_ih, const float* __restrict__ W_hh,
    const float* __restrict__ b_ih, const float* __restrict__ b_hh,
    const float* __restrict__ W1,   const float* __restrict__ b1)
{
    if (m < 40) {
        if (k < 10)  return W_hh[m * 10 + k];
        if (k == 10) return W_ih[m];            // W_ih is (40,1)
        return b_ih[m] + b_hh[m];               // k == 11 (bias column)
    } else if (m < 45) {
        int j = m - 40;
        if (k < 10)  return W1[j * 10 + k];
        if (k == 10) return 0.0f;
        return b1[j];                           // k == 11
    }
    return 0.0f;
}

__global__ __launch_bounds__(THREADS) void lstm_disc_kernel(
    const float* __restrict__ values, const float* __restrict__ masks,
    const float* __restrict__ W_ih,   const float* __restrict__ W_hh,
    const float* __restrict__ b_ih,   const float* __restrict__ b_hh,
    const float* __restrict__ W1,     const float* __restrict__ b1,
    const float* __restrict__ W2,     const float* __restrict__ b2,
    float* __restrict__ out, long long nBatch)
{
    __shared__ float lds[WAVES * WAVE_LDS];
    const int lane = threadIdx.x & 31;
    const int wave = threadIdx.x >> 5;
    float* xl = &lds[wave * WAVE_LDS + X_OFF];   // x values, recycled as score buffer
    float* hl = &lds[wave * WAVE_LDS + H_OFF];
    float* dl = &lds[wave * WAVE_LDS + D_OFF];

    const long long gw      = (long long)blockIdx.x * WAVES + wave;
    const long long myBatch = gw * 32 + lane;

    // Prefetch the mask bytes this lane will copy at the end (global_prefetch_b8):
    // pulls them toward L2 while the recurrent loop runs.
    __builtin_prefetch(masks + myBatch * TT, 0, 3);

    // ---- stage this lane's 20 input values into LDS (coalesced 80B chunk), zero h state
    {
        const float4* vs = (const float4*)(values + myBatch * TT);
        float4* xd = (float4*)(xl + lane * TT);
        #pragma unroll
        for (int i = 0; i < 5; ++i) xd[i] = vs[i];
        #pragma unroll
        for (int k = 0; k < 12; ++k) hl[lane * 12 + k] = 0.0f;
    }

    // ---- loop-invariant A fragments for V_WMMA_F32_16X16X4_F32
    // A 16x4 layout: lanes 0-15 hold M=lane, K={0,1}; lanes 16-31 hold M=lane-16, K={2,3}
    const int cl = lane & 15;
    const int hi = lane >> 4;
    v2f afr[3][3];
    #pragma unroll
    for (int mt = 0; mt < 3; ++mt) {
        #pragma unroll
        for (int kc = 0; kc < 3; ++kc) {
            const int m  = mt * 16 + cl;
            const int k0 = kc * 4 + hi * 2;
            afr[mt][kc][0] = waug(m, k0,     W_ih, W_hh, b_ih, b_hh, W1, b1);
            afr[mt][kc][1] = waug(m, k0 + 1, W_ih, W_hh, b_ih, b_hh, W1, b1);
        }
    }
    float w2r[5];
    #pragma unroll
    for (int j = 0; j < 5; ++j) w2r[j] = W2[j];
    const float b2v = b2[0];

    float c[HH];
    #pragma unroll
    for (int k = 0; k < HH; ++k) c[k] = 0.0f;

    asm volatile("s_wait_dscnt 0" ::: "memory");   // LDS staging visible wave-wide

    for (int t = 0; t < TT; ++t) {
        // ---- build B fragments (12 x 32, two 16-col N tiles)
        // B 4x16 layout: vgpr0 lanes0-15=row(4kc+0), lanes16-31=row(4kc+1); vgpr1 rows(4kc+2/3)
        v2f bfr[3][2];
        #pragma unroll
        for (int nt = 0; nt < 2; ++nt) {
            const int col = nt * 16 + cl;
            #pragma unroll
            for (int kc = 0; kc < 3; ++kc) {
                const int r0 = kc * 4 + hi;        // 0..9 always
                const int r1 = kc * 4 + 2 + hi;    // kc<2: 2..7 ; kc==2: 10 or 11
                bfr[kc][nt][0] = hl[col * 12 + r0];
                if (kc < 2) {
                    bfr[kc][nt][1] = hl[col * 12 + r1];
                } else {
                    // row 10 = x_t(col) for lanes 0-15, row 11 = 1.0 for lanes 16-31
                    const float xv = xl[col * TT + t];
                    bfr[kc][nt][1] = hi ? 1.0f : xv;
                }
            }
        }

        // ---- 18x V_WMMA_F32_16X16X4_F32 : D(48x32) = W_aug(48x12) * [h;x;1](12x32)
        v8f acc[3][2];
        #pragma unroll
        for (int mt = 0; mt < 3; ++mt) {
            #pragma unroll
            for (int nt = 0; nt < 2; ++nt) {
                v8f a = {};
                #pragma unroll
                for (int kc = 0; kc < 3; ++kc) {
                    a = __builtin_amdgcn_wmma_f32_16x16x4_f32(
                            false, afr[mt][kc], false, bfr[kc][nt],
                            (short)0, a, false, false);
                }
                acc[mt][nt] = a;
            }
        }

        // ---- scatter D to LDS (column-major per batch column)
        // D layout: vgpr i: lanes0-15 -> (M=i, N=lane); lanes16-31 -> (M=8+i, N=lane-16)
        #pragma unroll
        for (int mt = 0; mt < 3; ++mt) {
            #pragma unroll
            for (int nt = 0; nt < 2; ++nt) {
                const int col = nt * 16 + cl;
                const int rb  = mt * 16 + hi * 8;
                #pragma unroll
                for (int i = 0; i < 8; ++i)
                    dl[col * 48 + rb + i] = acc[mt][nt][i];
            }
        }
        asm volatile("s_wait_dscnt 0" ::: "memory");

        // ---- per-lane elementwise: lane owns batch column `lane`
        const float* gp = &dl[lane * 48];
        float sa = b2v;
        #pragma unroll
        for (int j = 0; j < 5; ++j) {
            const float v  = gp[40 + j];                 // W1*h + b1 (bias fused in GEMM)
            const float lr = v > 0.0f ? v : 0.2f * v;    // leaky_relu(0.2)
            sa = __builtin_fmaf(lr, w2r[j], sa);
        }
        // x value for step t was consumed above -> recycle its LDS slot as the
        // score buffer (flushed with coalesced float4 stores after the loop).
        xl[lane * TT + t] = sigmf(sa);                   // score uses entry-h: correct order

        #pragma unroll
        for (int k = 0; k < HH; ++k) {
            const float ig = sigmf(gp[k]);
            const float fg = sigmf(gp[10 + k]);
            const float gg = tanhfast(gp[20 + k]);
            const float og = sigmf(gp[30 + k]);
            c[k] = __builtin_fmaf(fg, c[k], ig * gg);
            hl[lane * 12 + k] = og * tanhfast(c[k]);
        }
        asm volatile("s_wait_dscnt 0" ::: "memory");     // h_new/scores visible for next step
    }

    // ---- flush scores (B,T) row-major: 5 coalesced float4 stores per lane
    {
        const float4* ss = (const float4*)(xl + lane * TT);
        float4* sd = (float4*)(out + myBatch * TT);
        #pragma unroll
        for (int i = 0; i < 5; ++i) sd[i] = ss[i];
    }

    // ---- pass-through masks output (float4, 80B per lane, aligned, prefetched)
    {
        const float4* ms = (const float4*)(masks + myBatch * TT);
        float4* md = (float4*)(out + nBatch * TT + myBatch * TT);
        #pragma unroll
        for (int i = 0; i < 5; ++i) md[i] = ms[i];
    }
}

extern "C" void kernel_launch(void* const* d_in, const int* in_sizes, int n_in,
                              void* d_out, int out_size, void* d_ws, size_t ws_size,
                              hipStream_t stream)
{
    const float* values = (const float*)d_in[0];
    const float* masks  = (const float*)d_in[1];
    const float* W_ih   = (const float*)d_in[2];
    const float* W_hh   = (const float*)d_in[3];
    const float* b_ih   = (const float*)d_in[4];
    const float* b_hh   = (const float*)d_in[5];
    const float* W1     = (const float*)d_in[6];
    const float* b1     = (const float*)d_in[7];
    const float* W2     = (const float*)d_in[8];
    const float* b2     = (const float*)d_in[9];
    float* out = (float*)d_out;

    const long long nBatch = (long long)in_sizes[0] / TT;     // 262144
    const int blocks = (int)(nBatch / (32 * WAVES));          // 2048 blocks * 4 waves * 32 batch
    lstm_disc_kernel<<<blocks, THREADS, 0, stream>>>(
        values, masks, W_ih, W_hh, b_ih, b_hh, W1, b1, W2, b2, out, nBatch);
}